// CtrlPointHungarianMatcher_1468878815836
// MI455X (gfx1250) — compile-verified
//
#include <hip/hip_runtime.h>
#include <hip/hip_bf16.h>
#include <math.h>

// ---------------- problem constants ----------------
#define BS     16
#define NQ     100
#define NPTS   25
#define VOC    96
#define MAXLEN 25
#define NGT    32
#define EDIM   300
#define ALPHA  0.25f
#define EPS_KL 1e-6f

#define NROW   (BS * NQ)            // 1600 queries
#define NCOL   (BS * NGT)           // 512 targets
#define SIMROWS (BS * NGT * MAXLEN) // 12800 gathered rows

typedef float v2f __attribute__((ext_vector_type(2)));
typedef float v8f __attribute__((ext_vector_type(8)));
union V8 { v8f v; float f[8]; };

// workspace layout (floats)
#define WS_CLSC 0                       // 1600
#define WS_LOGP (WS_CLSC + NROW)        // 1600*96
#define WS_SIM  (WS_LOGP + NROW * VOC)  // 12800*96
#define WS_TGTS (WS_SIM + SIMROWS * VOC)// 512*96
#define WS_NENT (WS_TGTS + NCOL * VOC)  // 512
#define WS_LEN  (WS_NENT + NCOL)        // 512
#define WS_KL   (WS_LEN + NCOL)         // 1600*32

__device__ __forceinline__ float wave_max(float v) {
#pragma unroll
  for (int m = 16; m >= 1; m >>= 1) v = fmaxf(v, __shfl_xor(v, m, 32));
  return v;
}
__device__ __forceinline__ float wave_sum(float v) {
#pragma unroll
  for (int m = 16; m >= 1; m >>= 1) v += __shfl_xor(v, m, 32);
  return v;
}

// ---------------- K1: focal classification cost ----------------
__global__ void k_class_cost(const float* __restrict__ pred_logits,
                             float* __restrict__ clsc) {
  int t = blockIdx.x * blockDim.x + threadIdx.x;
  if (t >= NROW) return;
  const float* p = pred_logits + t * NPTS;
  float s = 0.f;
#pragma unroll
  for (int i = 0; i < NPTS; ++i) s += 1.f / (1.f + __expf(-p[i]));
  float prob = s * (1.f / NPTS);
  float pos = ALPHA * (1.f - prob) * (1.f - prob) * (-logf(prob + 1e-8f));
  float neg = (1.f - ALPHA) * prob * prob * (-logf(1.f - prob + 1e-8f));
  clsc[t] = pos - neg;
}

// ---------------- K2: softmax over 97, mean over 25 pts -> logp ----------------
// one wave per (b,q)
__global__ void k_pred_logp(const float* __restrict__ text_logits,
                            float* __restrict__ logp) {
  int bq = blockIdx.x;            // 0..1599
  int lane = threadIdx.x;         // 0..31
  float a0 = 0.f, a1 = 0.f, a2 = 0.f;
  for (int pt = 0; pt < NPTS; ++pt) {
    const float* L = text_logits + (bq * NPTS + pt) * (VOC + 1);
    float x0 = L[lane];
    float x1 = L[lane + 32];
    float x2 = L[lane + 64];
    float x3 = (lane == 0) ? L[96] : -1e30f;
    float m = wave_max(fmaxf(fmaxf(x0, x1), fmaxf(x2, x3)));
    float e0 = __expf(x0 - m);
    float e1 = __expf(x1 - m);
    float e2 = __expf(x2 - m);
    float e3 = (lane == 0) ? __expf(x3 - m) : 0.f;
    float inv = 1.f / wave_sum(e0 + e1 + e2 + e3);
    a0 += e0 * inv; a1 += e1 * inv; a2 += e2 * inv;  // drop class VOC (pad)
  }
  const float r = 1.f / NPTS;
  float* o = logp + bq * VOC;
  o[lane]      = logf(fmaxf(a0 * r, EPS_KL));
  o[lane + 32] = logf(fmaxf(a1 * r, EPS_KL));
  o[lane + 64] = logf(fmaxf(a2 * r, EPS_KL));
}

// ---------------- K3: sim = gather(centroids)[12800x300] @ centroids^T[300x96] ----------------
// fp32 WMMA 16x16x4, one wave per 16x16 tile; 800 row tiles x 6 col tiles
__global__ void k_sim_wmma(const int* __restrict__ target_texts,
                           const float* __restrict__ centroids,
                           float* __restrict__ sim) {
  int tile = blockIdx.x;
  int rowTile = tile / (VOC / 16);
  int colTile = tile % (VOC / 16);
  int lane = threadIdx.x;
  int mi = lane & 15;
  int half = lane >> 4;

  // A row gather: flat sim row index == flat target_texts index
  int R = rowTile * 16 + mi;
  int ch = target_texts[R];
  ch = min(max(ch, 0), VOC - 1);
  const float* Arow = centroids + ch * EDIM;
  // B column: centroids^T -> B[k][n] = centroids[n][k]
  const float* Bcol = centroids + (colTile * 16 + mi) * EDIM;

  V8 acc; acc.v = (v8f){};
  for (int k0 = 0; k0 < EDIM; k0 += 4) {   // 75 iterations
    int kk = k0 + half * 2;
    float2 av = *(const float2*)(Arow + kk);
    float2 bv = *(const float2*)(Bcol + kk);
    v2f a = {av.x, av.y};
    v2f b = {bv.x, bv.y};
    acc.v = __builtin_amdgcn_wmma_f32_16x16x4_f32(
        false, a, false, b, (short)0, acc.v, false, false);
  }
  const float scale = rsqrtf((float)EDIM);
#pragma unroll
  for (int v = 0; v < 8; ++v) {
    int row = rowTile * 16 + v + half * 8;
    sim[row * VOC + colTile * 16 + mi] = acc.f[v] * scale;
  }
}

// ---------------- K4: per-(b,g) row softmax, masked mean, normalize, neg-entropy ----------------
// one 128-thread block (4 waves) per (b,g)
__global__ void k_tgt_stats(const int* __restrict__ target_texts,
                            const float* __restrict__ sim,
                            float* __restrict__ tgts,
                            float* __restrict__ nent,
                            float* __restrict__ lens) {
  int bg = blockIdx.x;             // 0..511
  int tid = threadIdx.x;
  int lane = tid & 31;
  int w = tid >> 5;
  __shared__ float colsum[VOC];
  __shared__ int cnt;
  if (tid < VOC) colsum[tid] = 0.f;
  if (tid == 0) cnt = 0;
  __syncthreads();

  for (int l = w; l < MAXLEN; l += 4) {
    bool valid = (target_texts[bg * MAXLEN + l] != VOC);
    const float* sp = sim + (bg * MAXLEN + l) * VOC;
    float x0 = sp[lane], x1 = sp[lane + 32], x2 = sp[lane + 64];
    float m = wave_max(fmaxf(fmaxf(x0, x1), x2));
    float e0 = __expf(x0 - m), e1 = __expf(x1 - m), e2 = __expf(x2 - m);
    float inv = 1.f / wave_sum(e0 + e1 + e2);
    if (valid) {
      atomicAdd(&colsum[lane],      e0 * inv);
      atomicAdd(&colsum[lane + 32], e1 * inv);
      atomicAdd(&colsum[lane + 64], e2 * inv);
      if (lane == 0) atomicAdd(&cnt, 1);
    }
  }
  __syncthreads();

  if (tid < 32) {
    float Lf = (float)cnt;
    float denom = 1.f / fmaxf(Lf, 1.f);
    float t0 = fmaxf(colsum[lane] * denom, EPS_KL);
    float t1 = fmaxf(colsum[lane + 32] * denom, EPS_KL);
    float t2 = fmaxf(colsum[lane + 64] * denom, EPS_KL);
    float T  = wave_sum(t0 + t1 + t2);
    float S2 = wave_sum(t0 * logf(t0) + t1 * logf(t1) + t2 * logf(t2));
    float invT = 1.f / T;
    float* o = tgts + bg * VOC;
    o[lane]      = t0 * invT;
    o[lane + 32] = t1 * invT;
    o[lane + 64] = t2 * invT;
    if (lane == 0) {
      nent[bg] = S2 * invT - logf(T);   // sum (t/T) log(t/T)
      lens[bg] = Lf;
    }
  }
}

// ---------------- K5: kl = relu(neg_ent - logp . tgt_s) via fp32 WMMA ----------------
// per batch b: (100x96) x (96x32); 7 row tiles x 2 col tiles per batch
__global__ void k_kl_wmma(const float* __restrict__ logp,
                          const float* __restrict__ tgts,
                          const float* __restrict__ nent,
                          const float* __restrict__ lens,
                          float* __restrict__ kl) {
  int id = blockIdx.x;             // 0..223
  int b = id / 14;
  int t = id % 14;
  int rowTile = t >> 1;            // 0..6
  int colTile = t & 1;             // 0..1
  int lane = threadIdx.x;
  int mi = lane & 15;
  int half = lane >> 4;

  int q = rowTile * 16 + mi;
  int qa = min(q, NQ - 1);                        // clamp pad rows (no EXEC change)
  const float* Arow = logp + (b * NQ + qa) * VOC;
  int g = colTile * 16 + mi;                      // 0..31
  const float* Bcol = tgts + (b * NGT + g) * VOC; // B[k][n] = tgt_s[g=n][k]

  V8 acc; acc.v = (v8f){};
  for (int k0 = 0; k0 < VOC; k0 += 4) {  // 24 iterations
    int kk = k0 + half * 2;
    float2 av = *(const float2*)(Arow + kk);
    float2 bv = *(const float2*)(Bcol + kk);
    v2f a = {av.x, av.y};
    v2f bb = {bv.x, bv.y};
    acc.v = __builtin_amdgcn_wmma_f32_16x16x4_f32(
        false, a, false, bb, (short)0, acc.v, false, false);
  }
#pragma unroll
  for (int v = 0; v < 8; ++v) {
    int qq = rowTile * 16 + v + half * 8;
    if (qq < NQ) {
      int gc = colTile * 16 + mi;
      float ne = nent[b * NGT + gc];
      float val = fmaxf(ne - acc.f[v], 0.f);
      if (lens[b * NGT + gc] == 0.f) val = 100.f;
      kl[(b * NQ + qq) * NGT + gc] = val;
    }
  }
}

// ---------------- K6: final assembly C = class + L1(cdist) + blockdiag(kl) ----------------
// one block per 16-row x 32-col output tile (col tile == one target batch b2)
__global__ void k_assemble(const float* __restrict__ pred_pts,
                           const float* __restrict__ tgt_pts,
                           const float* __restrict__ clsc,
                           const float* __restrict__ kl,
                           float* __restrict__ out) {
  int rt = blockIdx.x / BS;        // 0..99
  int b2 = blockIdx.x % BS;        // 0..15 (== col tile)
  int r0 = rt * 16;
  int c0 = b2 * NGT;
  int tid = threadIdx.x;

  __shared__ float qp[16 * 2 * NPTS];   // 800
  __shared__ float gp[NGT * 2 * NPTS];  // 1600
  for (int i = tid; i < 16 * 2 * NPTS; i += blockDim.x)
    qp[i] = pred_pts[(r0 + i / (2 * NPTS)) * (2 * NPTS) + i % (2 * NPTS)];
  for (int i = tid; i < NGT * 2 * NPTS; i += blockDim.x)
    gp[i] = tgt_pts[(c0 + i / (2 * NPTS)) * (2 * NPTS) + i % (2 * NPTS)];
  __syncthreads();

  for (int o = tid; o < 16 * NGT; o += blockDim.x) {
    int lr = o / NGT;
    int lc = o % NGT;
    int r = r0 + lr;               // = b*NQ + q
    float s = 0.f;
    const float* a = qp + lr * (2 * NPTS);
    const float* bb = gp + lc * (2 * NPTS);
#pragma unroll
    for (int d = 0; d < 2 * NPTS; ++d) s += fabsf(a[d] - bb[d]);
    float val = s + clsc[r];
    if (r / NQ == b2) val += kl[r * NGT + lc];
    out[r * NCOL + c0 + lc] = val;
  }
}

extern "C" void kernel_launch(void* const* d_in, const int* in_sizes, int n_in,
                              void* d_out, int out_size, void* d_ws, size_t ws_size,
                              hipStream_t stream) {
  const float* pred_logits  = (const float*)d_in[0];
  const float* pred_pts     = (const float*)d_in[1];
  const float* text_logits  = (const float*)d_in[2];
  const float* tgt_pts      = (const float*)d_in[3];
  const int*   target_texts = (const int*)d_in[4];
  const float* centroids    = (const float*)d_in[5];
  float* out = (float*)d_out;
  float* ws  = (float*)d_ws;

  float* clsc = ws + WS_CLSC;
  float* logp = ws + WS_LOGP;
  float* sim  = ws + WS_SIM;
  float* tgts = ws + WS_TGTS;
  float* nent = ws + WS_NENT;
  float* lens = ws + WS_LEN;
  float* kl   = ws + WS_KL;

  k_class_cost<<<(NROW + 255) / 256, 256, 0, stream>>>(pred_logits, clsc);
  k_pred_logp<<<NROW, 32, 0, stream>>>(text_logits, logp);
  k_sim_wmma<<<(SIMROWS / 16) * (VOC / 16), 32, 0, stream>>>(target_texts, centroids, sim);
  k_tgt_stats<<<NCOL, 128, 0, stream>>>(target_texts, sim, tgts, nent, lens);
  k_kl_wmma<<<BS * 7 * 2, 32, 0, stream>>>(logp, tgts, nent, lens, kl);
  k_assemble<<<(NROW / 16) * BS, 256, 0, stream>>>(pred_pts, tgt_pts, clsc, kl, out);
}